// Router_74964359184413
// MI455X (gfx1250) — compile-verified
//
#include <hip/hip_runtime.h>
#include <hip/hip_bf16.h>

#define N_TOKENS  16384
#define HIDDEN    4096
#define N_EXPERTS 64
#define TOP_K     8

typedef float v2f __attribute__((ext_vector_type(2)));
typedef float v8f __attribute__((ext_vector_type(8)));

// ---- GEMM tiling ----
#define CHUNK_K     64                  // K slice staged through LDS per iteration
#define NCHUNK      (HIDDEN / CHUNK_K)  // 64
#define XS_STRIDE   68                  // padded x-tile row stride (floats); rows 16B aligned
#define WTP_STRIDE  130                 // paired W^T rows: [k/2][expert][2] floats, 8B aligned

// Block: 128 threads = 4 wave32 (256 blocks -> fills the WGP array). Wave w
// computes a 16-token x 64-expert tile with V_WMMA_F32_16X16X4_F32 (full fp32,
// matching the reference). x (HBM-streamed) is double-buffered through
// registers; W (L2-resident, 1 MiB reused by all blocks) is prefetched one
// chunk ahead and copied global->LDS directly, keeping VGPR pressure low
// enough to avoid spills.
__global__ __launch_bounds__(128)
__attribute__((amdgpu_waves_per_eu(1)))
void router_gemm_kernel(
    const float* __restrict__ x,     // [N_TOKENS, HIDDEN]
    const float* __restrict__ W,     // [N_EXPERTS, HIDDEN]
    const float* __restrict__ bias,  // [N_EXPERTS]
    float* __restrict__ router)      // [N_TOKENS, N_EXPERTS]
{
  __shared__ float xs[64 * XS_STRIDE];               // 64 token rows x CHUNK_K   ~17.4 KB
  __shared__ float wt[(CHUNK_K / 2) * WTP_STRIDE];   // W^T chunk, k-paired       ~16.6 KB

  const int tid     = threadIdx.x;     // 0..127
  const int wave    = tid >> 5;        // 0..3
  const int lane    = tid & 31;
  const int halfSel = lane >> 4;       // 0: lanes 0-15, 1: lanes 16-31
  const int l16     = lane & 15;

  const int blockTok  = blockIdx.x * 64;
  const int tokenBase = blockTok + wave * 16;

  const int sRow = tid >> 4;           // 0..7   staging row group
  const int sC4  = tid & 15;           // float4 column within the 64-float chunk

  v8f zero = {};
  v8f acc[4];
#pragma unroll
  for (int t = 0; t < 4; ++t) acc[t] = zero;

  float4 xv[8];   // x staging: rows sRow + 8*j, j=0..7  (register double buffer)

  auto load_x_chunk = [&](int k0) {
#pragma unroll
    for (int j = 0; j < 8; ++j) {
      const int row = sRow + 8 * j;              // 0..63
      xv[j] = *(const float4*)(x + (size_t)(blockTok + row) * HIDDEN + k0 + sC4 * 4);
    }
  };

  auto store_x_chunk = [&]() {
#pragma unroll
    for (int j = 0; j < 8; ++j) {
      const int row = sRow + 8 * j;
      *(float4*)(&xs[row * XS_STRIDE + sC4 * 4]) = xv[j];
    }
  };

  // W chunk: global (L2/WGP$-hot after prefetch) -> LDS, k-paired transpose
  auto copy_w_chunk = [&](int k0) {
#pragma unroll
    for (int j = 0; j < 8; ++j) {
      const int e = sRow + 8 * j;                // 0..63
      float4 v = *(const float4*)(W + (size_t)e * HIDDEN + k0 + sC4 * 4);
      const int kk = sC4 * 4;                    // kk..kk+3, kk even
      wt[((kk >> 1) + 0) * WTP_STRIDE + e * 2 + 0] = v.x;   // k = kk
      wt[((kk >> 1) + 0) * WTP_STRIDE + e * 2 + 1] = v.y;   // k = kk+1
      wt[((kk >> 1) + 1) * WTP_STRIDE + e * 2 + 0] = v.z;   // k = kk+2
      wt[((kk >> 1) + 1) * WTP_STRIDE + e * 2 + 1] = v.w;   // k = kk+3
    }
  };

  auto prefetch_w_chunk = [&](int k0) {          // warm 16 KB: 128 threads x 128 B
    __builtin_prefetch(W + (size_t)(tid >> 1) * HIDDEN + k0 + (tid & 1) * 32, 0, 3);
  };

  auto prefetch_x_chunk = [&](int k0) {          // warm 16 KB of next-next x slab
    __builtin_prefetch(x + (size_t)(blockTok + (tid >> 1)) * HIDDEN + k0 + (tid & 1) * 32, 0, 0);
  };

  // prologue: stage chunk 0
  load_x_chunk(0);
  prefetch_w_chunk(0);
  store_x_chunk();
  copy_w_chunk(0);
  __syncthreads();

  const float* xw = &xs[(wave * 16 + l16) * XS_STRIDE];

  for (int c = 0; c < NCHUNK; ++c) {
    const bool more = (c + 1 < NCHUNK);
    if (more) {
      load_x_chunk((c + 1) * CHUNK_K);           // issue HBM loads; hide behind WMMAs
      prefetch_w_chunk((c + 1) * CHUNK_K);       // warm L2/WGP$ for the W copy
    }
    if (c + 2 < NCHUNK) prefetch_x_chunk((c + 2) * CHUNK_K);

    // ---- Compute: 16 K-steps of 4, 4 expert tiles each (64 WMMAs) ----
#pragma unroll
    for (int kk = 0; kk < CHUNK_K; kk += 4) {
      const int kb = kk + 2 * halfSel;           // even
      // A 16x4 fp32 fragment: lanes 0-15 -> K,K+1 ; lanes 16-31 -> K+2,K+3
      v2f a = *(const v2f*)(&xw[kb]);
#pragma unroll
      for (int t = 0; t < 4; ++t) {
        // B fragment: one aligned b64 -> natural even VGPR pair
        v2f bf = *(const v2f*)(&wt[(kb >> 1) * WTP_STRIDE + (t * 16 + l16) * 2]);
        acc[t] = __builtin_amdgcn_wmma_f32_16x16x4_f32(
            /*neg_a=*/false, a, /*neg_b=*/false, bf,
            /*c_mod=*/(short)0, acc[t], /*reuse_a=*/false, /*reuse_b=*/false);
      }
    }

    if (more) {
      __syncthreads();                           // all waves done reading LDS
      store_x_chunk();                           // commit staged x chunk
      copy_w_chunk((c + 1) * CHUNK_K);           // short-latency L2 hit -> LDS
      __syncthreads();                           // LDS ready
    }
  }

  // ---- Epilogue: add bias, store logits (C/D layout: VGPR i -> rows i / i+8) ----
#pragma unroll
  for (int t = 0; t < 4; ++t) {
    const int   col = t * 16 + l16;
    const float bv  = bias[col];
#pragma unroll
    for (int i = 0; i < 8; ++i) {
      const int row = tokenBase + i + 8 * halfSel;
      router[(size_t)row * N_EXPERTS + col] = acc[t][i] + bv;
    }
  }
}

// One thread per token: top-8 of 64 logits (ties: lower index first, matching
// jax.lax.top_k), renormalized softmax over the selected 8 (mathematically equal
// to softmax-then-renormalize), one-hot mask written [E, K, N] (coalesced in N).
__global__ __launch_bounds__(256) void router_topk_kernel(
    const float* __restrict__ router,   // [N_TOKENS, N_EXPERTS]
    float* __restrict__ weights,        // [N_TOKENS, TOP_K]
    float* __restrict__ indices,        // [N_TOKENS, TOP_K] (integer-valued)
    float* __restrict__ mask)           // [N_EXPERTS, TOP_K, N_TOKENS] (0/1)
{
  const int n = blockIdx.x * blockDim.x + threadIdx.x;
  const float* r = router + (size_t)n * N_EXPERTS;

  float lg[N_EXPERTS];
#pragma unroll
  for (int e = 0; e < N_EXPERTS; ++e) lg[e] = r[e];

  float selv[TOP_K];
  int   seli[TOP_K];
  float pv = 3.0e38f;
  int   pi = -1;
#pragma unroll
  for (int kk = 0; kk < TOP_K; ++kk) {
    float bv = -3.0e38f;
    int   bi = 0;
#pragma unroll
    for (int e = 0; e < N_EXPERTS; ++e) {
      // still a candidate: strictly after (pv,pi) in (value desc, index asc) order
      const bool cand   = (lg[e] < pv) || (lg[e] == pv && e > pi);
      const bool better = cand && (lg[e] > bv);
      if (better) { bv = lg[e]; bi = e; }
    }
    selv[kk] = bv;
    seli[kk] = bi;
    pv = bv;
    pi = bi;
  }

  // softmax over the selected 8 logits == renormalized top-k softmax probs
  const float m = selv[0];
  float ex[TOP_K];
  float s = 0.0f;
#pragma unroll
  for (int kk = 0; kk < TOP_K; ++kk) {
    ex[kk] = __expf(selv[kk] - m);
    s += ex[kk];
  }
  const float inv = 1.0f / s;

#pragma unroll
  for (int kk = 0; kk < TOP_K; ++kk) {
    weights[(size_t)n * TOP_K + kk] = ex[kk] * inv;
    indices[(size_t)n * TOP_K + kk] = (float)seli[kk];
  }

  // expert_mask[e][kk][n]
  for (int e = 0; e < N_EXPERTS; ++e) {
#pragma unroll
    for (int kk = 0; kk < TOP_K; ++kk) {
      mask[((size_t)e * TOP_K + kk) * N_TOKENS + n] = (seli[kk] == e) ? 1.0f : 0.0f;
    }
  }
}

extern "C" void kernel_launch(void* const* d_in, const int* in_sizes, int n_in,
                              void* d_out, int out_size, void* d_ws, size_t ws_size,
                              hipStream_t stream) {
  const float* x    = (const float*)d_in[0];
  const float* W    = (const float*)d_in[1];
  const float* bias = (const float*)d_in[2];

  float* out     = (float*)d_out;
  float* router  = out;                                        // [16384, 64]
  float* weights = router  + (size_t)N_TOKENS * N_EXPERTS;     // [16384, 8]
  float* indices = weights + (size_t)N_TOKENS * TOP_K;         // [16384, 8]
  float* mask    = indices + (size_t)N_TOKENS * TOP_K;         // [64, 8, 16384]

  router_gemm_kernel<<<N_TOKENS / 64, 128, 0, stream>>>(x, W, bias, router);
  router_topk_kernel<<<N_TOKENS / 256, 256, 0, stream>>>(router, weights, indices, mask);
}